// FusedEPMoE_30777735643467
// MI455X (gfx1250) — compile-verified
//
#include <hip/hip_runtime.h>

// MoE: T tokens, H hidden, F intermediate, E experts, top-K routing.
#define T_TOK 1024
#define HID   2048
#define FDIM  1024
#define NEXP  8
#define TOPK  2

#define TM  32   // token rows per block tile
#define TN  64   // output cols per block tile
#define TKC 32   // K-chunk staged through LDS

#define WPITCH 68  // LDS pitch for 64-wide tiles (64 + 4 pad dwords)
#define APITCH 36  // LDS pitch for 32-wide tiles (32 + 4 pad dwords)

typedef __attribute__((ext_vector_type(2))) float v2f;
typedef __attribute__((ext_vector_type(8))) float v8f;
typedef __attribute__((ext_vector_type(4))) unsigned int v4u;
typedef __attribute__((ext_vector_type(8))) int v8i_;
typedef __attribute__((ext_vector_type(4))) int v4i_;

// ---------------------------------------------------------------- TDM helper
// Build a D# for a 2D tile load (data_size = 4B) and issue tensor_load_to_lds.
//   g1d0 packs data_size + LDS pad config (see CDNA5 D# group1 bit layout).
//   OOB rows/cols (vs td0/td1) are zero-filled by the TDM.
__device__ __forceinline__ void tdm_load_tile_2d(const void* gptr, unsigned lds_off,
                                                 unsigned g1d0,
                                                 unsigned td0, unsigned td1,
                                                 unsigned tile0, unsigned tile1,
                                                 unsigned stride0) {
  unsigned long long ga = (unsigned long long)(uintptr_t)gptr;
  v4u g0;
  g0[0] = 1u;                                                 // count=1, user D#
  g0[1] = lds_off;                                            // lds_addr (bytes)
  g0[2] = (unsigned)ga;                                       // global_addr[31:0]
  g0[3] = ((unsigned)(ga >> 32) & 0x01FFFFFFu) | (2u << 30);  // addr[56:32] | type=2
  v8i_ g1;
  g1[0] = (int)g1d0;                                          // mask=0|dsize|pad cfg
  g1[1] = (int)((td0 & 0xFFFFu) << 16);                       // tensor_dim0[15:0]
  g1[2] = (int)((td0 >> 16) | ((td1 & 0xFFFFu) << 16));       // td0[31:16]|td1[15:0]
  g1[3] = (int)((td1 >> 16) | (tile0 << 16));                 // td1[31:16]|tile_dim0
  g1[4] = (int)tile1;                                         // tile_dim1 (tile_dim2=0)
  g1[5] = (int)stride0;                                       // tensor_dim0_stride
  g1[6] = 0;
  g1[7] = 0;
  v4i_ z4 = {0, 0, 0, 0};                                     // groups 2/3 unused (2D)
  v8i_ z8 = {0, 0, 0, 0, 0, 0, 0, 0};
  // 6-arg toolchain form: (g0, g1, g2, g3, extra_group, cpol)
  __builtin_amdgcn_tensor_load_to_lds(g0, g1, z4, z4, z8, 0);
}

// group1 dword0 configs: data_size=4B (code 2), pad_enable=1
//   64-dword rows + 4 pad dwords: pad_interval code 5 (64), pad_amount code 3 (4)
#define G1D0_P64 ((2u << 16) | (1u << 20) | (5u << 22) | (3u << 25))
//   32-dword rows + 4 pad dwords: pad_interval code 4 (32), pad_amount code 3 (4)
#define G1D0_P32 ((2u << 16) | (1u << 20) | (4u << 22) | (3u << 25))

__device__ __forceinline__ unsigned lds_offset(const void* p) {
  return (unsigned)(uintptr_t)p;  // generic LDS pointer: low 32 bits = LDS offset
}

// ---------------------------------------------------------------- zero init
__global__ __launch_bounds__(256)
void moe_zero_kernel(float* __restrict__ out, int n, int* __restrict__ counts) {
  int idx = blockIdx.x * blockDim.x + threadIdx.x;
  int stride = gridDim.x * blockDim.x;
  for (int i = idx; i < n; i += stride) out[i] = 0.0f;
  if (blockIdx.x == 0 && threadIdx.x < NEXP) counts[threadIdx.x] = 0;
}

// ---------------------------------------------------------------- router
__global__ __launch_bounds__(256)
void moe_router_kernel(const float* __restrict__ logits,
                       int* __restrict__ tok, float* __restrict__ wl,
                       int* __restrict__ counts) {
  int t = blockIdx.x * blockDim.x + threadIdx.x;
  if (t >= T_TOK) return;
  float l[NEXP];
  float mx = -1e30f;
#pragma unroll
  for (int e = 0; e < NEXP; ++e) { l[e] = logits[t * NEXP + e]; mx = fmaxf(mx, l[e]); }
#pragma unroll
  for (int e = 0; e < NEXP; ++e) l[e] = __expf(l[e] - mx);
  int i0 = 0; float p0 = -1.0f;
#pragma unroll
  for (int e = 0; e < NEXP; ++e) if (l[e] > p0) { p0 = l[e]; i0 = e; }
  int i1 = 0; float p1 = -1.0f;
#pragma unroll
  for (int e = 0; e < NEXP; ++e) if (e != i0 && l[e] > p1) { p1 = l[e]; i1 = e; }
  float inv = 1.0f / (p0 + p1);
  int pos0 = atomicAdd(&counts[i0], 1);
  tok[i0 * T_TOK + pos0] = t; wl[i0 * T_TOK + pos0] = p0 * inv;
  int pos1 = atomicAdd(&counts[i1], 1);
  tok[i1 * T_TOK + pos1] = t; wl[i1 * T_TOK + pos1] = p1 * inv;
}

// ---------------------------------------------------------------- scan
__global__ void moe_scan_kernel(const int* __restrict__ counts, int* __restrict__ offsets) {
  if (threadIdx.x == 0) {
    int acc = 0;
#pragma unroll
    for (int e = 0; e < NEXP; ++e) { offsets[e] = acc; acc += counts[e]; }
  }
}

// ---------------------------------------------------------------- gate+up GEMM
// act[slot, f] = routeW * silu(x·w1) * (x·w3), per expert over its token list.
// X tile (gathered rows) staged manually; W1/W3 tiles streamed via TDM.
__global__ __launch_bounds__(256)
void moe_gateup_kernel(const float* __restrict__ x,
                       const float* __restrict__ w1, const float* __restrict__ w3,
                       const int* __restrict__ tok, const float* __restrict__ wl,
                       const int* __restrict__ counts, const int* __restrict__ offsets,
                       float* __restrict__ act) {
  const int e    = blockIdx.x / (T_TOK / TM);
  const int tile = blockIdx.x % (T_TOK / TM);
  const int cnt  = counts[e];
  const int m0   = tile * TM;
  if (m0 >= cnt) return;
  const int fbase = blockIdx.y * TN;
  const int off   = offsets[e];

  __shared__ __align__(16) float Xs[TM * APITCH];
  __shared__ __align__(16) float W1s[TKC * WPITCH];
  __shared__ __align__(16) float W3s[TKC * WPITCH];
  __shared__ int toks[TM];

  const int tid  = threadIdx.x;
  const int lane = tid & 31;
  const int wv   = tid >> 5;
  const int mi   = wv >> 2;   // 0..1 : 16-row subtile
  const int ni   = wv & 3;    // 0..3 : 16-col subtile

  if (tid < TM) {
    int p  = m0 + tid;
    int pc = (p < cnt) ? p : (cnt - 1);   // clamp: never read garbage list slots
    toks[tid] = tok[e * T_TOK + pc];
  }
  __syncthreads();

  const float* w1e = w1 + (size_t)e * HID * FDIM;
  const float* w3e = w3 + (size_t)e * HID * FDIM;
  const unsigned w1s_off = lds_offset(W1s);
  const unsigned w3s_off = lds_offset(W3s);

  v8f accG = {};
  v8f accU = {};

  const int arow = (mi << 4) + (lane & 15);
  const int bcol = (ni << 4) + (lane & 15);
  const int ksel = (lane >> 4) << 1;  // lanes 0-15: K=0,1 ; lanes 16-31: K=2,3

  for (int kc = 0; kc < HID; kc += TKC) {
    // Wave 0 issues TDM loads of the two weight tiles (wave-uniform D#).
    if (wv == 0) {
      tdm_load_tile_2d(w1e + (size_t)kc * FDIM + fbase, w1s_off, G1D0_P64,
                       FDIM, HID, TN, TKC, FDIM);
      tdm_load_tile_2d(w3e + (size_t)kc * FDIM + fbase, w3s_off, G1D0_P64,
                       FDIM, HID, TN, TKC, FDIM);
    }
    // All threads stage the gathered X tile (TM x TKC).
    for (int i = tid; i < TM * TKC; i += 256) {
      int r = i >> 5, c = i & 31;
      Xs[r * APITCH + c] = x[(size_t)toks[r] * HID + kc + c];
    }
    __builtin_amdgcn_s_wait_tensorcnt(0);  // no-op for waves with TENSORcnt==0
    __syncthreads();
#pragma unroll
    for (int kk = 0; kk < TKC; kk += 4) {
      int kb = kk + ksel;
      v2f a;  a[0]  = Xs[arow * APITCH + kb];  a[1]  = Xs[arow * APITCH + kb + 1];
      v2f b1; b1[0] = W1s[kb * WPITCH + bcol]; b1[1] = W1s[(kb + 1) * WPITCH + bcol];
      v2f b3; b3[0] = W3s[kb * WPITCH + bcol]; b3[1] = W3s[(kb + 1) * WPITCH + bcol];
      accG = __builtin_amdgcn_wmma_f32_16x16x4_f32(false, a, false, b1,
                                                   (short)0, accG, false, false);
      accU = __builtin_amdgcn_wmma_f32_16x16x4_f32(false, a, false, b3,
                                                   (short)0, accU, false, false);
    }
    __syncthreads();
  }

  // epilogue: silu(g)*u, scaled by routing weight, into compacted act buffer
  const int colg = fbase + (ni << 4) + (lane & 15);
#pragma unroll
  for (int r = 0; r < 8; ++r) {
    int m = (mi << 4) + r + ((lane >> 4) << 3);
    int p = m0 + m;
    if (p < cnt) {
      float g  = accG[r];
      float u  = accU[r];
      float sg = g * (1.0f / (1.0f + __expf(-g)));
      float wr = wl[e * T_TOK + p];
      act[(size_t)(off + p) * FDIM + colg] = sg * u * wr;
    }
  }
}

// ---------------------------------------------------------------- down GEMM
// out[t, h] += act[slot, :] · w2_e[:, h]. Both tiles streamed via TDM
// (act rows are compacted -> regular 2D tile; TDM zero-fills OOB rows).
__global__ __launch_bounds__(256)
void moe_down_kernel(const float* __restrict__ w2, const float* __restrict__ act,
                     const int* __restrict__ tok, const int* __restrict__ counts,
                     const int* __restrict__ offsets, float* __restrict__ out) {
  const int e    = blockIdx.x / (T_TOK / TM);
  const int tile = blockIdx.x % (T_TOK / TM);
  const int cnt  = counts[e];
  const int m0   = tile * TM;
  if (m0 >= cnt) return;
  const int hbase = blockIdx.y * TN;
  const int off   = offsets[e];

  __shared__ __align__(16) float As[TM * APITCH];
  __shared__ __align__(16) float Ws[TKC * WPITCH];

  const int tid  = threadIdx.x;
  const int lane = tid & 31;
  const int wv   = tid >> 5;
  const int mi   = wv >> 2;
  const int ni   = wv & 3;

  const float* w2e = w2 + (size_t)e * FDIM * HID;
  const unsigned as_off = lds_offset(As);
  const unsigned ws_off = lds_offset(Ws);

  v8f acc = {};

  const int arow = (mi << 4) + (lane & 15);
  const int bcol = (ni << 4) + (lane & 15);
  const int ksel = (lane >> 4) << 1;

  for (int kc = 0; kc < FDIM; kc += TKC) {
    if (wv == 0) {
      // act tile: rows off+m0.., cols kc.. ; tensor is (T*K) x F, OOB rows -> 0
      tdm_load_tile_2d(act + (size_t)(off + m0) * FDIM + kc, as_off, G1D0_P32,
                       FDIM, T_TOK * TOPK, TKC, TM, FDIM);
      // w2 tile: rows kc.., cols hbase..
      tdm_load_tile_2d(w2e + (size_t)kc * HID + hbase, ws_off, G1D0_P64,
                       HID, FDIM, TN, TKC, HID);
    }
    __builtin_amdgcn_s_wait_tensorcnt(0);
    __syncthreads();
#pragma unroll
    for (int kk = 0; kk < TKC; kk += 4) {
      int kb = kk + ksel;
      v2f a; a[0] = As[arow * APITCH + kb];  a[1] = As[arow * APITCH + kb + 1];
      v2f b; b[0] = Ws[kb * WPITCH + bcol];  b[1] = Ws[(kb + 1) * WPITCH + bcol];
      acc = __builtin_amdgcn_wmma_f32_16x16x4_f32(false, a, false, b,
                                                  (short)0, acc, false, false);
    }
    __syncthreads();
  }

  const int colh = hbase + (ni << 4) + (lane & 15);
#pragma unroll
  for (int r = 0; r < 8; ++r) {
    int m = (mi << 4) + r + ((lane >> 4) << 3);
    int p = m0 + m;
    if (p < cnt) {
      int t = tok[e * T_TOK + p];
      atomicAdd(&out[(size_t)t * HID + colh], acc[r]);
    }
  }
}

// ---------------------------------------------------------------- launch
extern "C" void kernel_launch(void* const* d_in, const int* in_sizes, int n_in,
                              void* d_out, int out_size, void* d_ws, size_t ws_size,
                              hipStream_t stream) {
  const float* x  = (const float*)d_in[0];  // [T, H]
  const float* rl = (const float*)d_in[1];  // [T, E]
  const float* w1 = (const float*)d_in[2];  // [E, H, F]
  const float* w3 = (const float*)d_in[3];  // [E, H, F]
  const float* w2 = (const float*)d_in[4];  // [E, F, H]
  float* out = (float*)d_out;               // [T, H]

  // workspace layout
  float* act     = (float*)d_ws;                                   // (T*K) x F
  int*   tok     = (int*)(act + (size_t)T_TOK * TOPK * FDIM);      // E*T
  float* wl      = (float*)(tok + NEXP * T_TOK);                   // E*T
  int*   counts  = (int*)(wl + NEXP * T_TOK);                      // E
  int*   offsets = counts + NEXP;                                  // E

  moe_zero_kernel<<<1024, 256, 0, stream>>>(out, T_TOK * HID, counts);
  moe_router_kernel<<<(T_TOK + 255) / 256, 256, 0, stream>>>(rl, tok, wl, counts);
  moe_scan_kernel<<<1, 32, 0, stream>>>(counts, offsets);

  dim3 g1(NEXP * (T_TOK / TM), FDIM / TN);
  moe_gateup_kernel<<<g1, 256, 0, stream>>>(x, w1, w3, tok, wl, counts, offsets, act);

  dim3 g2(NEXP * (T_TOK / TM), HID / TN);
  moe_down_kernel<<<g2, 256, 0, stream>>>(w2, act, tok, counts, offsets, out);
}